// SurfaceDensity_75222057222756
// MI455X (gfx1250) — compile-verified
//
#include <hip/hip_runtime.h>

// SurfaceDensity histogram for MI455X (gfx1250, wave32).
// Memory-bound: ~256 MB streamed once -> ~11 us floor at 23.3 TB/s.
// v2: 4 particles/thread/iter via three global_load_b128 (48B, always
// 16B-aligned) + one b128 mass load, all non-temporal (stream > L2);
// per-lane padded LDS sub-histograms (stride 21, gcd(21,64)=1 => conflict-free
// ds_add_f32 across a wave32); deterministic two-stage reduction.

typedef float v4f __attribute__((ext_vector_type(4)));

#define SD_R_BINS   20
#define SD_INV_DR   2.0f      // 1/DR, DR = 0.5 (exact power-of-two multiply)
#define SD_PAD      21        // 20 bins + 1 pad float per lane row
#define SD_BLOCK    256       // 8 wave32s per block
#define SD_GRID     2048
#define SD_FCHUNKS  32        // finalize parallel chunk groups

__global__ void __launch_bounds__(SD_BLOCK)
sd_hist_partial(const float* __restrict__ pos,    // [N,3] interleaved f32
                const float* __restrict__ mass,   // [N] f32
                float* __restrict__ partial,      // [gridDim.x * SD_R_BINS]
                int n)
{
    __shared__ float lds[SD_BLOCK * SD_PAD];      // 21504 B / block
    const int tid = threadIdx.x;
    float* row = &lds[tid * SD_PAD];

    #pragma unroll
    for (int b = 0; b < SD_PAD; ++b) row[b] = 0.0f;
    __syncthreads();

    const long long T   = (long long)blockDim.x * gridDim.x;
    const long long gid = (long long)blockIdx.x * blockDim.x + tid;
    const long long nchunks = (long long)(n >> 2);   // 4 particles per chunk

    const v4f* __restrict__ p4 = (const v4f*)pos;    // 48B per chunk, 16B aligned
    const v4f* __restrict__ m4 = (const v4f*)mass;

    for (long long j = gid; j < nchunks; j += T) {
        const v4f A = __builtin_nontemporal_load(&p4[3 * j]);     // x0 y0 z0 x1
        const v4f B = __builtin_nontemporal_load(&p4[3 * j + 1]); // y1 z1 x2 y2
        const v4f C = __builtin_nontemporal_load(&p4[3 * j + 2]); // z2 x3 y3 z3
        const v4f M = __builtin_nontemporal_load(&m4[j]);         // m0..m3

        const long long nj = j + T;                  // stream-ahead hint
        if (nj < nchunks) __builtin_prefetch(&p4[3 * nj], 0, 1);

        const float xs[4] = {A.x, A.w, B.z, C.y};
        const float ys[4] = {A.y, B.x, B.w, C.z};
        const float ms[4] = {M.x, M.y, M.z, M.w};

        #pragma unroll
        for (int k = 0; k < 4; ++k) {
            const float r = sqrtf(xs[k] * xs[k] + ys[k] * ys[k]);
            const int   b = (int)(r * SD_INV_DR);    // trunc == floor, r >= 0
            if (b < SD_R_BINS) atomicAdd(&row[b], ms[k]);  // ds_add_f32, lane-private
        }
    }

    // Scalar tail for n % 4 != 0 (not taken for N = 16M).
    for (long long i = (nchunks << 2) + gid; i < n; i += T) {
        const float x = pos[3 * i], y = pos[3 * i + 1], m = mass[i];
        const float r = sqrtf(x * x + y * y);
        const int   b = (int)(r * SD_INV_DR);
        if (b < SD_R_BINS) atomicAdd(&row[b], m);
    }
    __syncthreads();

    // Deterministic log2 tree fold of the 256 rows into row 0.
    for (int off = SD_BLOCK / 2; off > 0; off >>= 1) {
        if (tid < off) {
            float*       a = &lds[tid * SD_PAD];
            const float* c = &lds[(tid + off) * SD_PAD];
            #pragma unroll
            for (int b = 0; b < SD_R_BINS; ++b) a[b] += c[b];
        }
        __syncthreads();
    }

    if (tid < SD_R_BINS)
        partial[(long long)blockIdx.x * SD_R_BINS + tid] = lds[tid];
}

__global__ void __launch_bounds__(SD_FCHUNKS * SD_R_BINS)
sd_finalize(const float* __restrict__ partial,    // [nblocks * SD_R_BINS]
            const float* __restrict__ area,       // [SD_R_BINS]
            float* __restrict__ out,              // [SD_R_BINS]
            int nblocks)
{
    __shared__ float fsum[SD_FCHUNKS * SD_R_BINS];
    const int tid = threadIdx.x;                  // 0..639
    const int b   = tid % SD_R_BINS;
    const int c   = tid / SD_R_BINS;

    float s = 0.0f;
    for (int g = c; g < nblocks; g += SD_FCHUNKS) // fixed order -> deterministic
        s += partial[(long long)g * SD_R_BINS + b];
    fsum[tid] = s;
    __syncthreads();

    if (tid < SD_R_BINS) {
        float t = 0.0f;
        #pragma unroll
        for (int c2 = 0; c2 < SD_FCHUNKS; ++c2)   // fixed order fold
            t += fsum[c2 * SD_R_BINS + tid];
        out[tid] = t / area[tid];
    }
}

extern "C" void kernel_launch(void* const* d_in, const int* in_sizes, int n_in,
                              void* d_out, int out_size, void* d_ws, size_t ws_size,
                              hipStream_t stream)
{
    const float* pos  = (const float*)d_in[0];   // positions [N,3] f32
    const float* mass = (const float*)d_in[1];   // masses    [N]   f32
    const float* area = (const float*)d_in[2];   // area      [20]  f32
    const int n = in_sizes[1];                   // N (masses element count)

    int grid = SD_GRID;
    const int ws_blocks = (int)(ws_size / (SD_R_BINS * sizeof(float)));
    if (grid > ws_blocks) grid = ws_blocks;
    if (grid < 1) grid = 1;

    float* partial = (float*)d_ws;

    sd_hist_partial<<<grid, SD_BLOCK, 0, stream>>>(pos, mass, partial, n);
    sd_finalize<<<1, SD_FCHUNKS * SD_R_BINS, 0, stream>>>(partial, area,
                                                          (float*)d_out, grid);
}